// SGD_MRVGAE_15625091022923
// MI455X (gfx1250) — compile-verified
//
#include <hip/hip_runtime.h>
#include <math.h>

#define N_NODES 100000
#define EG      800000
#define NE      262144
#define D_IN    128
#define D_H     256
#define D_Zl    256
#define CATS    4
#define D_M     64
#define D_DEC   256
#define D_OUTF  128
#define EPSF    1e-7f

typedef __attribute__((ext_vector_type(2))) float v2f;
typedef __attribute__((ext_vector_type(8))) float v8f;

// ---------------- degree / norm ----------------
__global__ void deg_kernel(const int* __restrict__ src, const int* __restrict__ dst,
                           float* __restrict__ deg_out, float* __restrict__ deg_in, int ne) {
  int e = blockIdx.x * blockDim.x + threadIdx.x;
  if (e >= ne) return;
  atomicAdd(&deg_out[src[e]], 1.0f);
  atomicAdd(&deg_in[dst[e]], 1.0f);
}

__global__ void norm_kernel(const float* __restrict__ deg_out, const float* __restrict__ deg_in,
                            float* __restrict__ norm_out, float* __restrict__ norm_in, int n) {
  int i = blockIdx.x * blockDim.x + threadIdx.x;
  if (i >= n) return;
  float a = deg_out[i];
  float b = deg_in[i];
  norm_out[i] = (a > 0.0f) ? (1.0f / sqrtf(a)) : 1.0f;
  norm_in[i]  = (b > 0.0f) ? (1.0f / sqrtf(b)) : 1.0f;
}

// ------------- scatter: agg[dst] += x[src]*norm_out[src] -------------
__global__ void scatter_kernel(const float* __restrict__ x, const float* __restrict__ norm_out,
                               const int* __restrict__ src, const int* __restrict__ dst,
                               float* __restrict__ agg, int D) {
  long tid = (long)blockIdx.x * blockDim.x + threadIdx.x;
  int per = D >> 2;
  long tot = (long)EG * per;
  if (tid >= tot) return;
  int e = (int)(tid / per);
  int f = (int)(tid % per) << 2;
  int s = src[e], d = dst[e];
  float ns = norm_out[s];
  float4 v = *(const float4*)(x + (size_t)s * D + f);
  float* o = agg + (size_t)d * D + f;
  atomicAdd(o + 0, v.x * ns);
  atomicAdd(o + 1, v.y * ns);
  atomicAdd(o + 2, v.z * ns);
  atomicAdd(o + 3, v.w * ns);
}

// ------------- WMMA f32 GEMM: C = act(op(A) @ W + bias) -------------
// Block tile 64(M) x 64(N), K-chunk 32. 8 waves; each wave owns TWO adjacent
// 16x16 output tiles sharing one A fragment. LDS layouts chosen so every
// fragment is one aligned ds_load_b64 into an even VGPR pair:
//   As  [64][36] row-major  (36: b64-aligned reads, b128-aligned stores)
//   BsT [64][34] transposed (34: (B[k][n],B[k+1][n]) contiguous + aligned)
// GATHER:   A_row(g) = A[gsrc[g]] + A[gdst[g]]   (edge u_add_v fused into load)
// ROWSCALE: A_row(g) *= rowscale[g]              (folds D^-1/2 into GEMM)
// ACT: 0 = none, 1 = relu
template <bool GATHER, bool ROWSCALE, int ACT>
__global__ void __launch_bounds__(256)
gemm_wmma(const float* __restrict__ A, const float* __restrict__ W,
          const float* __restrict__ bias, float* __restrict__ C,
          const int* __restrict__ gsrc, const int* __restrict__ gdst,
          const float* __restrict__ rowscale, int M, int K, int N) {
  __shared__ float As[64][36];
  __shared__ float BsT[64][34];
  int tid  = threadIdx.x;
  int lane = tid & 31;
  int wave = tid >> 5;
  int half = lane >> 4;
  int l16  = lane & 15;
  int msub = wave & 3;        // 4 row tiles of 16
  int nsub = wave >> 2;       // 2 col groups of 32 (2 tiles each)
  int m0 = blockIdx.y * 64;
  int n0 = blockIdx.x * 64;

  v8f acc0 = {0.f, 0.f, 0.f, 0.f, 0.f, 0.f, 0.f, 0.f};
  v8f acc1 = {0.f, 0.f, 0.f, 0.f, 0.f, 0.f, 0.f, 0.f};

  for (int k0 = 0; k0 < K; k0 += 32) {
    // stage A tile 64x32 (float4 per thread, 2 iterations; b128 LDS stores)
#pragma unroll
    for (int it = 0; it < 2; ++it) {
      int idx = it * 256 + tid;
      int r = idx >> 3;
      int c = (idx & 7) << 2;
      int g = m0 + r;
      float4 v = make_float4(0.f, 0.f, 0.f, 0.f);
      if (g < M) {
        if (GATHER) {
          int s = gsrc[g], d = gdst[g];
          float4 va = *(const float4*)(A + (size_t)s * K + k0 + c);
          float4 vb = *(const float4*)(A + (size_t)d * K + k0 + c);
          v = make_float4(va.x + vb.x, va.y + vb.y, va.z + vb.z, va.w + vb.w);
        } else {
          v = *(const float4*)(A + (size_t)g * K + k0 + c);
          if (ROWSCALE) {
            float rs = rowscale[g];
            v.x *= rs; v.y *= rs; v.z *= rs; v.w *= rs;
          }
        }
      }
      *(float4*)&As[r][c] = v;
    }
    // stage B tile 32x64 transposed into LDS (coalesced global reads)
#pragma unroll
    for (int it = 0; it < 8; ++it) {
      int idx = it * 256 + tid;
      int kk = idx >> 6;
      int n  = idx & 63;
      float v = 0.f;
      if (n0 + n < N) v = W[(size_t)(k0 + kk) * N + n0 + n];
      BsT[n][kk] = v;
    }
    __syncthreads();

    int arow  = msub * 16 + l16;
    int bcol0 = nsub * 32 + l16;
#pragma unroll
    for (int kk = 0; kk < 32; kk += 4) {
      int ka = kk + half * 2;
      v2f a  = *(const v2f*)&As[arow][ka];
      v2f b0 = *(const v2f*)&BsT[bcol0][ka];
      v2f b1 = *(const v2f*)&BsT[bcol0 + 16][ka];
      acc0 = __builtin_amdgcn_wmma_f32_16x16x4_f32(false, a, false, b0, (short)0, acc0,
                                                   false, false);
      acc1 = __builtin_amdgcn_wmma_f32_16x16x4_f32(false, a, false, b1, (short)0, acc1,
                                                   false, false);
    }
    __syncthreads();
  }

  // epilogue: two 16x16 tiles per wave
#pragma unroll
  for (int t = 0; t < 2; ++t) {
    int gcol = n0 + nsub * 32 + t * 16 + l16;
    if (gcol < N) {
      float bv = bias ? bias[gcol] : 0.f;
#pragma unroll
      for (int i = 0; i < 8; ++i) {
        int grow = m0 + msub * 16 + half * 8 + i;
        if (grow < M) {
          float v = (t == 0 ? acc0[i] : acc1[i]) + bv;
          if (ACT == 1) v = v > 0.f ? v : 0.f;
          C[(size_t)grow * N + gcol] = v;
        }
      }
    }
  }
}

// ------------- fused reparam + gumbel-softmax mixture + A-softmax -------------
// one wave (32 lanes) per edge; lane handles dims d = lane, lane+32 of M[E,64]
__global__ void __launch_bounds__(256)
fuse_branch(const float* __restrict__ mean, const float* __restrict__ logstd,
            const float* __restrict__ q, const float* __restrict__ gauss,
            const float* __restrict__ unif, const float* __restrict__ Wc,
            const float* __restrict__ bc, float* __restrict__ Mout,
            float* __restrict__ Aout, const int* __restrict__ temp_ptr) {
  int lane = threadIdx.x & 31;
  int wave = threadIdx.x >> 5;
  int e = blockIdx.x * 8 + wave;
  if (e >= NE) return;

  // decode temp (python scalar: int-valued or float bit pattern)
  int iv = *temp_ptr;
  float temp = (iv > 0 && iv < (1 << 23)) ? (float)iv : __int_as_float(iv);

  // Gumbel-softmax Z over CATS categories (redundant per lane, cheap)
  float lg[CATS];
  float mx = -3.4e38f;
#pragma unroll
  for (int c = 0; c < CATS; ++c) {
    float u = unif[(size_t)e * CATS + c];
    float gum = -logf(-logf(u + EPSF) + EPSF);
    float l = (q[(size_t)e * CATS + c] + gum) / temp;
    lg[c] = l;
    mx = fmaxf(mx, l);
  }
  float se = 0.f;
#pragma unroll
  for (int c = 0; c < CATS; ++c) {
    lg[c] = expf(lg[c] - mx);
    se += lg[c];
  }
  float inv = 1.f / se;

  // M[e,d] = sum_c Z_c * (gauss*exp(logstd) + mean)
  size_t base = (size_t)e * D_Zl;
  float m0 = 0.f, m1 = 0.f;
#pragma unroll
  for (int c = 0; c < CATS; ++c) {
    float z = lg[c] * inv;
    int o0 = c * D_M + lane;
    int o1 = o0 + 32;
    m0 += z * (gauss[base + o0] * expf(logstd[base + o0]) + mean[base + o0]);
    m1 += z * (gauss[base + o1] * expf(logstd[base + o1]) + mean[base + o1]);
  }
  Mout[(size_t)e * D_M + lane] = m0;
  Mout[(size_t)e * D_M + lane + 32] = m1;

  // A = softmax(M @ Wc + bc) ; Wc is [64, CATS]
  float p[CATS];
#pragma unroll
  for (int j = 0; j < CATS; ++j)
    p[j] = m0 * Wc[lane * CATS + j] + m1 * Wc[(lane + 32) * CATS + j];
#pragma unroll
  for (int off = 16; off > 0; off >>= 1) {
#pragma unroll
    for (int j = 0; j < CATS; ++j) p[j] += __shfl_xor(p[j], off, 32);
  }
  if (lane < CATS) {
    float s[CATS];
    float mx2 = -3.4e38f;
#pragma unroll
    for (int j = 0; j < CATS; ++j) {
      s[j] = p[j] + bc[j];
      mx2 = fmaxf(mx2, s[j]);
    }
    float tot = 0.f;
#pragma unroll
    for (int j = 0; j < CATS; ++j) tot += expf(s[j] - mx2);
    Aout[(size_t)e * CATS + lane] = expf(s[lane] - mx2) / tot;
  }
}

extern "C" void kernel_launch(void* const* d_in, const int* in_sizes, int n_in,
                              void* d_out, int out_size, void* d_ws, size_t ws_size,
                              hipStream_t stream) {
  (void)in_sizes; (void)n_in; (void)out_size; (void)ws_size;
  const float* x        = (const float*)d_in[0];
  const int*   src      = (const int*)d_in[1];
  const int*   dst      = (const int*)d_in[2];
  const int*   pos_src  = (const int*)d_in[3];
  const int*   pos_dst  = (const int*)d_in[4];
  const int*   neg_src  = (const int*)d_in[5];
  const int*   neg_dst  = (const int*)d_in[6];
  const float* gauss_pos= (const float*)d_in[7];
  const float* gauss_neg= (const float*)d_in[8];
  const float* unif_pos = (const float*)d_in[9];
  const float* unif_neg = (const float*)d_in[10];
  const int*   temp_ptr = (const int*)d_in[11];
  const float* W1 = (const float*)d_in[12]; const float* b1 = (const float*)d_in[13];
  const float* W2 = (const float*)d_in[14]; const float* b2 = (const float*)d_in[15];
  const float* Wm = (const float*)d_in[16]; const float* bm = (const float*)d_in[17];
  const float* Wl = (const float*)d_in[18]; const float* bl = (const float*)d_in[19];
  const float* Wq = (const float*)d_in[20]; const float* bq = (const float*)d_in[21];
  const float* Wd1= (const float*)d_in[22]; const float* bd1= (const float*)d_in[23];
  const float* WdX= (const float*)d_in[24]; const float* bdX= (const float*)d_in[25];
  const float* Wc = (const float*)d_in[26]; const float* bc = (const float*)d_in[27];

  // workspace layout (floats)
  float* wsf = (float*)d_ws;
  size_t o = 0;
  float* deg_out  = wsf + o; o += N_NODES;
  float* deg_in   = wsf + o; o += N_NODES;
  float* norm_out = wsf + o; o += N_NODES;
  float* norm_in  = wsf + o; o += N_NODES;
  float* agg  = wsf + o; o += (size_t)N_NODES * D_H;
  float* h1   = wsf + o; o += (size_t)N_NODES * D_H;
  float* h2   = wsf + o; o += (size_t)N_NODES * D_H;
  float* Mbuf = wsf + o; o += (size_t)NE * D_M;
  float* Dbuf = wsf + o; o += (size_t)NE * D_DEC;

  // output layout (flat concat in reference return order)
  float* out = (float*)d_out;
  const size_t E = NE;
  float* posA       = out;
  float* negA       = posA + E * CATS;
  float* posX       = negA + E * CATS;
  float* negX       = posX + E * D_OUTF;
  float* pos_mean   = negX + E * D_OUTF;
  float* neg_mean   = pos_mean + E * D_Zl;
  float* pos_logstd = neg_mean + E * D_Zl;
  float* neg_logstd = pos_logstd + E * D_Zl;
  float* posq       = neg_logstd + E * D_Zl;
  float* negq       = posq + E * CATS;

  // ---- degrees & norms ----
  hipMemsetAsync(deg_out, 0, 2 * (size_t)N_NODES * sizeof(float), stream);
  deg_kernel<<<(EG + 255) / 256, 256, 0, stream>>>(src, dst, deg_out, deg_in, EG);
  norm_kernel<<<(N_NODES + 255) / 256, 256, 0, stream>>>(deg_out, deg_in, norm_out, norm_in,
                                                         N_NODES);

  // ---- GraphConv layer 1: relu((D_in^-1/2 * scatter(x * D_out^-1/2)) @ W1 + b1) ----
  hipMemsetAsync(agg, 0, (size_t)N_NODES * D_IN * sizeof(float), stream);
  {
    long tot = (long)EG * (D_IN / 4);
    scatter_kernel<<<(unsigned)((tot + 255) / 256), 256, 0, stream>>>(x, norm_out, src, dst,
                                                                      agg, D_IN);
  }
  gemm_wmma<false, true, 1><<<dim3((D_H + 63) / 64, (N_NODES + 63) / 64), 256, 0, stream>>>(
      agg, W1, b1, h1, nullptr, nullptr, norm_in, N_NODES, D_IN, D_H);

  // ---- GraphConv layer 2 (no relu) ----
  hipMemsetAsync(agg, 0, (size_t)N_NODES * D_H * sizeof(float), stream);
  {
    long tot = (long)EG * (D_H / 4);
    scatter_kernel<<<(unsigned)((tot + 255) / 256), 256, 0, stream>>>(h1, norm_out, src, dst,
                                                                      agg, D_H);
  }
  gemm_wmma<false, true, 0><<<dim3((D_H + 63) / 64, (N_NODES + 63) / 64), 256, 0, stream>>>(
      agg, W2, b2, h2, nullptr, nullptr, norm_in, N_NODES, D_H, D_H);

  // ---- per-branch decoder (npemb gather fused into GEMM A-loads) ----
  dim3 gE256((D_Zl + 63) / 64, NE / 64);
  for (int b = 0; b < 2; ++b) {
    const int*   es    = b ? neg_src   : pos_src;
    const int*   ed    = b ? neg_dst   : pos_dst;
    const float* gauss = b ? gauss_neg : gauss_pos;
    const float* unif  = b ? unif_neg  : unif_pos;
    float* omean   = b ? neg_mean   : pos_mean;
    float* ologstd = b ? neg_logstd : pos_logstd;
    float* oq      = b ? negq : posq;
    float* oA      = b ? negA : posA;
    float* oX      = b ? negX : posX;

    gemm_wmma<true, false, 0><<<gE256, 256, 0, stream>>>(h2, Wm, bm, omean, es, ed, nullptr,
                                                         NE, D_H, D_Zl);
    gemm_wmma<true, false, 0><<<gE256, 256, 0, stream>>>(h2, Wl, bl, ologstd, es, ed, nullptr,
                                                         NE, D_H, D_Zl);
    gemm_wmma<true, false, 0><<<dim3(1, NE / 64), 256, 0, stream>>>(h2, Wq, bq, oq, es, ed,
                                                                    nullptr, NE, D_H, CATS);
    fuse_branch<<<NE / 8, 256, 0, stream>>>(omean, ologstd, oq, gauss, unif, Wc, bc, Mbuf, oA,
                                            temp_ptr);
    gemm_wmma<false, false, 1><<<dim3((D_DEC + 63) / 64, NE / 64), 256, 0, stream>>>(
        Mbuf, Wd1, bd1, Dbuf, nullptr, nullptr, nullptr, NE, D_M, D_DEC);
    gemm_wmma<false, false, 1><<<dim3((D_OUTF + 63) / 64, NE / 64), 256, 0, stream>>>(
        Dbuf, WdX, bdX, oX, nullptr, nullptr, nullptr, NE, D_DEC, D_OUTF);
  }
}